// Seq2Seq_67473936220306
// MI455X (gfx1250) — compile-verified
//
#include <hip/hip_runtime.h>
#include <hip/hip_bf16.h>

typedef float v2f __attribute__((ext_vector_type(2)));
typedef float v8f __attribute__((ext_vector_type(8)));

#define HID 20
#define EMB 768
#define TDEC 285
#define G4 80            // 4*HID gate width
#define CHUNK 64         // scan prefetch chunk (steps)

// ---------- fast activations (v_exp_f32 / v_rcp_f32) ----------
__device__ __forceinline__ float fsig(float x) {
    return __builtin_amdgcn_rcpf(1.0f + __builtin_amdgcn_exp2f(-1.44269504088896f * x));
}
__device__ __forceinline__ float ftanh(float x) {
    float e = __builtin_amdgcn_exp2f(2.88539008177793f * x);
    return 1.0f - 2.0f * __builtin_amdgcn_rcpf(e + 1.0f);
}

// ---------- CDNA5 async global->LDS helpers ----------
__device__ __forceinline__ void async_copy_b128(unsigned ldsAddr, const void* gaddr) {
    // each lane copies 16 bytes from the global address in vaddr to the LDS address in vdst  (tracked by ASYNCcnt)
    asm volatile("global_load_async_to_lds_b128 %0, %1, off"
                 :: "v"(ldsAddr), "v"(gaddr) : "memory");
}
__device__ __forceinline__ void wait_async0() {
#if __has_builtin(__builtin_amdgcn_s_wait_asynccnt)
    __builtin_amdgcn_s_wait_asynccnt(0);
#else
    asm volatile("s_wait_asynccnt 0x0" ::: "memory");
#endif
}

// =====================================================================
// GEMM: Y[M x 80] = X[M x 768] @ W[80 x 768]^T + bias[80]
// fp32 WMMA 16x16x4. One wave per 16-row strip, 5 N-tiles (80 cols).
// FULL=true: M is a multiple of the strip size -> no edge guards in the
// hot loop (removes per-iteration EXEC masking seen in round-1 asm).
// =====================================================================
template <bool FULL>
__global__ __launch_bounds__(256) void gemm_in80(
    const float* __restrict__ X, const float* __restrict__ W,
    const float* __restrict__ bias, float* __restrict__ Y, int M)
{
    const int lane = threadIdx.x & 31;
    const int wave = threadIdx.x >> 5;
    const int row0 = (blockIdx.x * 8 + wave) * 16;
    if (!FULL && row0 >= M) return;
    const int half = lane >> 4;   // which K-half of the frag this lane holds
    const int l16  = lane & 15;
    const int aRow = row0 + l16;
    const bool aIn = FULL || (aRow < M);
    const int K = EMB;

    v8f acc[5] = {};
    for (int kk = 0; kk < K; kk += 4) {
        const int kO = kk + 2 * half;
        v2f a;
        if (aIn) { a.x = X[aRow * K + kO]; a.y = X[aRow * K + kO + 1]; }
        else     { a.x = 0.0f;             a.y = 0.0f; }
#pragma unroll
        for (int n = 0; n < 5; ++n) {
            const int col = n * 16 + l16;
            v2f b;
            b.x = W[col * K + kO];       // B[k][col] = W[col][k]
            b.y = W[col * K + kO + 1];
            acc[n] = __builtin_amdgcn_wmma_f32_16x16x4_f32(
                false, a, false, b, (short)0, acc[n], false, false);
        }
    }
#pragma unroll
    for (int n = 0; n < 5; ++n) {
        const int col = n * 16 + l16;
        const float bv = bias[col];
#pragma unroll
        for (int r = 0; r < 8; ++r) {
            const int row = row0 + r + 8 * half;
            if (FULL || row < M) Y[row * G4 + col] = acc[n][r] + bv;
        }
    }
}

// =====================================================================
// Prep: Kmat[m][j] = sum_e Wfc[e][m] * Wih_d[j][e]   (20 x 80)
//       k0b[j]    = bih_d[j] + sum_e bfc[e] * Wih_d[j][e]
// =====================================================================
__global__ __launch_bounds__(256) void prep_kernel(
    const float* __restrict__ Wfc, const float* __restrict__ Wih_d,
    const float* __restrict__ bih_d, const float* __restrict__ bfc,
    float* __restrict__ Kmat, float* __restrict__ k0b)
{
    int t = blockIdx.x * 256 + threadIdx.x;
    if (t < HID * G4) {
        int m = t / G4, j = t % G4;
        float s = 0.0f;
        for (int e = 0; e < EMB; ++e) s += Wfc[e * HID + m] * Wih_d[j * EMB + e];
        Kmat[t] = s;
    } else if (t < HID * G4 + G4) {
        int j = t - HID * G4;
        float s = bih_d[j];
        for (int e = 0; e < EMB; ++e) s += bfc[e] * Wih_d[j * EMB + e];
        k0b[j] = s;
    }
}

// =====================================================================
// Sequential scan: encoder (32768 steps) then decoder (284 steps).
// 1 block x 128 threads (4 waves). Lane k < 20 owns hidden unit k.
// xg / gxLab are prefetched via CDNA5 async global->LDS copies into a
// double buffer: chunk c+1 is issued BEFORE computing chunk c, and
// s_wait_asynccnt + barrier land after the compute, hiding HBM latency
// behind the serial LSTM chain without burning staging VGPRs.
// =====================================================================
__global__ __launch_bounds__(128) void scan_kernel(
    const float* __restrict__ xg, const float* __restrict__ gxLab,
    const float* __restrict__ Kmat, const float* __restrict__ k0b,
    const float* __restrict__ Whh_e, const float* __restrict__ bhh_e,
    const float* __restrict__ Whh_d, const float* __restrict__ bhh_d,
    const int* __restrict__ tf_mask, float* __restrict__ Hbuf)
{
    __shared__ __attribute__((aligned(16))) float sbuf[2][CHUNK * G4];
    __shared__ int tf_sh[TDEC];

    const int tid  = threadIdx.x;
    const int lane = tid & 31;
    const int wave = tid >> 5;
    const int k = lane < HID ? lane : HID - 1;   // lanes 20..31 shadow lane 19

    for (int i = tid; i < TDEC; i += 128) tf_sh[i] = tf_mask[i];

    const unsigned ldsBuf0 = (unsigned)(uintptr_t)(&sbuf[0][0]);
    const unsigned ldsBuf1 = (unsigned)(uintptr_t)(&sbuf[1][0]);

    // issue async loads for one chunk: 5120 floats = 1280 x b128, 10 per thread
    auto stage = [&](const float* src, long baseFloat, long limitFloat, int buf) {
        const unsigned ldsBase = (buf ? ldsBuf1 : ldsBuf0) + (unsigned)tid * 16u;
#pragma unroll
        for (int i = 0; i < 10; ++i) {
            long f = baseFloat + (long)(tid + i * 128) * 4;   // start float idx
            if (f < limitFloat)
                async_copy_b128(ldsBase + (unsigned)i * (128u * 16u), src + f);
        }
    };

    // encoder recurrent weights (gate order i,f,g,o = rows k, k+20, k+40, k+60)
    float wE[4][HID], bE[4];
#pragma unroll
    for (int q = 0; q < 4; ++q) {
        bE[q] = bhh_e[k + HID * q];
#pragma unroll
        for (int m = 0; m < HID; ++m) wE[q][m] = Whh_e[(k + HID * q) * HID + m];
    }

    float h = 0.0f, c = 0.0f;
    float hv[HID];
#pragma unroll
    for (int m = 0; m < HID; ++m) hv[m] = 0.0f;

    // ---------------- encoder: 512 chunks of 64 steps ----------------
    const long encLimit = (long)32768 * G4;
    stage(xg, 0, encLimit, 0);
    wait_async0();
    __syncthreads();
    for (int cx = 0; cx < 512; ++cx) {
        if (cx + 1 < 512) stage(xg, (long)(cx + 1) * (CHUNK * G4), encLimit, (cx + 1) & 1);
        const float* xs = sbuf[cx & 1];
#pragma unroll 1
        for (int s = 0; s < CHUNK; ++s) {
            const float* gr = xs + s * G4;
            float gi = gr[k]           + bE[0];
            float gf = gr[k + HID]     + bE[1];
            float gg = gr[k + 2 * HID] + bE[2];
            float go = gr[k + 3 * HID] + bE[3];
#pragma unroll
            for (int m = 0; m < HID; ++m) {
                gi += hv[m] * wE[0][m];
                gf += hv[m] * wE[1][m];
                gg += hv[m] * wE[2][m];
                go += hv[m] * wE[3][m];
            }
            c = fsig(gf) * c + fsig(gi) * ftanh(gg);
            h = fsig(go) * ftanh(c);
#pragma unroll
            for (int m = 0; m < HID; ++m) hv[m] = __shfl(h, m, 32);
        }
        if (cx + 1 < 512) wait_async0();
        __syncthreads();
    }

    // ---------------- decoder: 284 steps ----------------
    float wD[4][HID], bD[4], kc[4][HID], k04[4];
#pragma unroll
    for (int q = 0; q < 4; ++q) {
        bD[q]  = bhh_d[k + HID * q];
        k04[q] = k0b[k + HID * q];
#pragma unroll
        for (int m = 0; m < HID; ++m) {
            wD[q][m] = Whh_d[(k + HID * q) * HID + m];
            kc[q][m] = Kmat[m * G4 + k + HID * q];
        }
    }
    const long decLimit = (long)TDEC * G4;
    stage(gxLab, 0, decLimit, 0);
    wait_async0();
    __syncthreads();
    const int nDec = (284 + CHUNK - 1) / CHUNK;   // 5
    for (int cx = 0; cx < nDec; ++cx) {
        if (cx + 1 < nDec) stage(gxLab, (long)(cx + 1) * (CHUNK * G4), decLimit, (cx + 1) & 1);
        const float* xs = sbuf[cx & 1];
        const int rem = 284 - cx * CHUNK;
        const int sEnd = rem < CHUNK ? rem : CHUNK;
#pragma unroll 1
        for (int s = 0; s < sEnd; ++s) {
            const int gs = cx * CHUNK + s;
            const float* gr = xs + s * G4;
            const int tf = (gs == 0) ? 1 : tf_sh[gs];
            float gi, gf, gg, go;
            if (tf > 0) {   // teacher forcing: gx = label projection (has bih_d)
                gi = gr[k];           gf = gr[k + HID];
                gg = gr[k + 2 * HID]; go = gr[k + 3 * HID];
            } else {        // gx = pred@Wih_d.T folded: h_{t-1} @ Kmat + k0
                gi = k04[0]; gf = k04[1]; gg = k04[2]; go = k04[3];
#pragma unroll
                for (int m = 0; m < HID; ++m) {
                    gi += hv[m] * kc[0][m];
                    gf += hv[m] * kc[1][m];
                    gg += hv[m] * kc[2][m];
                    go += hv[m] * kc[3][m];
                }
            }
            gi += bD[0]; gf += bD[1]; gg += bD[2]; go += bD[3];
#pragma unroll
            for (int m = 0; m < HID; ++m) {
                gi += hv[m] * wD[0][m];
                gf += hv[m] * wD[1][m];
                gg += hv[m] * wD[2][m];
                go += hv[m] * wD[3][m];
            }
            c = fsig(gf) * c + fsig(gi) * ftanh(gg);
            h = fsig(go) * ftanh(c);
#pragma unroll
            for (int m = 0; m < HID; ++m) hv[m] = __shfl(h, m, 32);
            if (wave == 0 && lane < HID) Hbuf[gs * HID + lane] = h;
        }
        if (cx + 1 < nDec) wait_async0();
        __syncthreads();
    }
}

// =====================================================================
// Output projection: out[285 x 768]; row0 = 0, row r = H[r-1] @ Wfc.T + bfc
// fp32 WMMA, one wave per 16x16 tile. 18 M-tiles x 48 N-tiles = 864 tiles.
// =====================================================================
__global__ __launch_bounds__(256) void outproj(
    const float* __restrict__ H, const float* __restrict__ Wfc,
    const float* __restrict__ bfc, float* __restrict__ out)
{
    const int lane = threadIdx.x & 31;
    const int wave = threadIdx.x >> 5;
    const int tile = blockIdx.x * 8 + wave;
    if (tile >= 18 * 48) return;
    const int mT = tile / 48, nT = tile % 48;
    const int row0 = mT * 16, col0 = nT * 16;
    const int half = lane >> 4, l16 = lane & 15;
    const int aRow = row0 + l16;

    v8f acc = {};
#pragma unroll
    for (int kk = 0; kk < HID; kk += 4) {
        const int kO = kk + 2 * half;
        v2f a;
        if (aRow >= 1 && aRow < TDEC) {
            a.x = H[(aRow - 1) * HID + kO];
            a.y = H[(aRow - 1) * HID + kO + 1];
        } else { a.x = 0.0f; a.y = 0.0f; }
        const int col = col0 + l16;
        v2f b;
        b.x = Wfc[col * HID + kO];      // B[k][col] = Wfc[col][k]
        b.y = Wfc[col * HID + kO + 1];
        acc = __builtin_amdgcn_wmma_f32_16x16x4_f32(
            false, a, false, b, (short)0, acc, false, false);
    }
    const int col = col0 + l16;
    const float bv = bfc[col];
#pragma unroll
    for (int r = 0; r < 8; ++r) {
        const int row = row0 + r + 8 * half;
        if (row < TDEC) out[row * EMB + col] = (row == 0) ? 0.0f : (acc[r] + bv);
    }
}

// =====================================================================
extern "C" void kernel_launch(void* const* d_in, const int* in_sizes, int n_in,
                              void* d_out, int out_size, void* d_ws, size_t ws_size,
                              hipStream_t stream) {
    const float* x     = (const float*)d_in[0];
    const float* label = (const float*)d_in[1];
    const int*   tf    = (const int*)  d_in[2];
    const float* Wih_e = (const float*)d_in[3];
    const float* Whh_e = (const float*)d_in[4];
    const float* bih_e = (const float*)d_in[5];
    const float* bhh_e = (const float*)d_in[6];
    const float* Wih_d = (const float*)d_in[7];
    const float* Whh_d = (const float*)d_in[8];
    const float* bih_d = (const float*)d_in[9];
    const float* bhh_d = (const float*)d_in[10];
    const float* Wfc   = (const float*)d_in[11];
    const float* bfc   = (const float*)d_in[12];

    float* ws    = (float*)d_ws;
    float* xg    = ws;                         // 32768*80 = 2,621,440 f
    float* gxLab = xg + 32768 * G4;            // 285*80   = 22,800 f
    float* Kmat  = gxLab + TDEC * G4;          // 1,600 f
    float* k0b   = Kmat + HID * G4;            // 80 f
    float* Hbuf  = k0b + G4;                   // 284*20 = 5,680 f

    // 1) encoder input projection (memory-bound WMMA GEMM, 96 MB stream)
    gemm_in80<true><<<256, 256, 0, stream>>>(x, Wih_e, bih_e, xg, 32768);
    // 2) decoder label projection (edge-guarded variant)
    gemm_in80<false><<<3, 256, 0, stream>>>(label, Wih_d, bih_d, gxLab, TDEC);
    // 3) fold Wfc into Wih_d for the non-teacher-forced path
    prep_kernel<<<7, 256, 0, stream>>>(Wfc, Wih_d, bih_d, bfc, Kmat, k0b);
    // 4) sequential encoder+decoder scan (latency-bound, async-LDS prefetch)
    scan_kernel<<<1, 128, 0, stream>>>(xg, gxLab, Kmat, k0b,
                                       Whh_e, bhh_e, Whh_d, bhh_d, tf, Hbuf);
    // 5) output projection (WMMA) + zero row 0
    outproj<<<108, 256, 0, stream>>>(Hbuf, Wfc, bfc, (float*)d_out);
}